// LegacyHexPlaneField_22514218566152
// MI455X (gfx1250) — compile-verified
//
#include <hip/hip_runtime.h>

// HexPlane field eval, MI455X (gfx1250, wave32).
// Strategy: 1 wave == 1 point, lane == channel (OUT_DIM==32==wave32).
//  - bilinear coords/weights wave-uniform; gather base forced to SGPRs via
//    readfirstlane -> global_load_b64 v, v_laneOff, s[base] offset:imm
//  - per-lane channel offsets (lane*H*W*4) hoisted out of the point loop
//  - all 36 gathers per point issued before consumption (deep MLP vs L2)
//  - planes (~56MB) stay L2-resident; output streamed with NT stores so the
//    384MB write stream doesn't evict the plane working set from L2
//  - CDNA5 paths: async global->LDS staging (ASYNCcnt), global_prefetch_b8,
//    split wait counters.

#define WAVES_PER_BLOCK 8
#define BLOCK_THREADS   (WAVES_PER_BLOCK * 32)
#define ASYNC_STAGE     1

struct PlanePtrs { const float* p[18]; };   // [res][plane] row-major, 3x6

typedef float v2f __attribute__((ext_vector_type(2), aligned(4)));

static __device__ __forceinline__ int imin(int a, int b) { return a < b ? a : b; }

__global__ void __launch_bounds__(BLOCK_THREADS)
hexplane_kernel(const float* __restrict__ pts,      // [N,3]
                const float* __restrict__ tstamps,  // [N]
                PlanePtrs g,
                float* __restrict__ out,            // [N,96]
                int N)
{
    __shared__ __align__(16) float sP[WAVES_PER_BLOCK * 4];  // x,y,z,t per staged point

    const int lane   = threadIdx.x & 31;
    const int wave   = threadIdx.x >> 5;
    const int stride = (int)gridDim.x * WAVES_PER_BLOCK;

    // Per-lane channel byte offsets, loop-invariant: [res][0=spatial SxS, 1=time 150xS]
    unsigned chOff[3][2];
    #pragma unroll
    for (int r = 0; r < 3; ++r) {
        const unsigned S = 64u << r;
        chOff[r][0] = (unsigned)lane * S * S * 4u;
        chOff[r][1] = (unsigned)lane * 150u * S * 4u;
    }

    for (int base = (int)blockIdx.x * WAVES_PER_BLOCK; base < N; base += stride) {
        // ---- stage 8 points' (x,y,z,t) into LDS; wave 0, one lane per float ----
        if (wave == 0) {
            const float* gaddr;
            unsigned lidx;
            if (lane < 24) {                       // 8 points x 3 coords
                int p = lane / 3;
                int c = lane - 3 * p;
                int idx = imin(base + p, N - 1);
                gaddr = pts + (size_t)idx * 3 + c;
                lidx  = (unsigned)(p * 4 + c);
            } else {                               // 8 timestamps
                int p = lane - 24;
                int idx = imin(base + p, N - 1);
                gaddr = tstamps + idx;
                lidx  = (unsigned)(p * 4 + 3);
            }
#if ASYNC_STAGE
            unsigned laddr = (unsigned)(unsigned long long)
                ((__attribute__((address_space(3))) float*)sP + lidx);
            asm volatile("global_load_async_to_lds_b32 %0, %1, off"
                         :: "v"(laddr), "v"(gaddr) : "memory");
            asm volatile("s_wait_asynccnt 0" ::: "memory");
#else
            sP[lidx] = *gaddr;
#endif
            // prefetch next chunk's coords (speculative: silently dropped if OOB)
            __builtin_prefetch(pts + (size_t)(base + stride) * 3, 0, 1);
            __builtin_prefetch(tstamps + (size_t)(base + stride), 0, 1);
        }
        __syncthreads();

        const int i = base + wave;
        if (i < N) {
            float4 P = *(const float4*)&sP[wave * 4];
            // world -> normalized: (x - B)*(2/(-2B)) - 1 == -0.625*x for B=1.6
            const float px  = P.x * -0.625f;
            const float py  = P.y * -0.625f;
            const float pz  = P.z * -0.625f;
            const float ptm = P.w;                 // time already in [-1,1]

            const char* sbase[18];
            unsigned    loff [18];
            unsigned    rowB [18];
            float       wxa  [18], wya[18];

            // ---- address + weight setup for all 18 planes ----
            #pragma unroll
            for (int r = 0; r < 3; ++r) {
                const int S = 64 << r;
                #pragma unroll
                for (int q = 0; q < 6; ++q) {
                    const int idx    = r * 6 + q;
                    const int isTime = (q == 2 || q >= 4) ? 1 : 0;
                    const int W = S;
                    const int H = isTime ? 150 : S;
                    const float cx = (q < 3) ? px : ((q < 5) ? py : pz);
                    const float cy = (q == 0) ? py
                                   : (q == 1 || q == 3) ? pz
                                   : ptm;

                    float xp = (cx + 1.0f) * (0.5f * (float)(W - 1));
                    float yp = (cy + 1.0f) * (0.5f * (float)(H - 1));
                    xp = fminf(fmaxf(xp, 0.0f), (float)(W - 1));
                    yp = fminf(fmaxf(yp, 0.0f), (float)(H - 1));
                    // border-equivalent: clamp to W-2 and let wx -> 1 select last texel
                    int x0 = imin((int)xp, W - 2);
                    int y0 = imin((int)yp, H - 2);
                    wxa[idx] = xp - (float)x0;
                    wya[idx] = yp - (float)y0;

                    // wave-uniform tile offset -> force into SGPRs (saddr form)
                    unsigned bo = (unsigned)((y0 * W + x0) * 4);
                    bo = (unsigned)__builtin_amdgcn_readfirstlane((int)bo);
                    sbase[idx] = (const char*)g.p[idx] + bo;
                    loff [idx] = chOff[r][isTime];
                    rowB [idx] = (unsigned)(W * 4);   // second-row immediate
                }
            }

            // ---- issue all 36 gathers (b64 pairs), then consume ----
            v2f r0[18], r1[18];
            #pragma unroll
            for (int idx = 0; idx < 18; ++idx) {
                r0[idx] = *(const v2f*)(sbase[idx] + loff[idx]);
                r1[idx] = *(const v2f*)(sbase[idx] + loff[idx] + rowB[idx]);
            }

            #pragma unroll
            for (int r = 0; r < 3; ++r) {
                float acc = 1.0f;
                #pragma unroll
                for (int q = 0; q < 6; ++q) {
                    const int idx = r * 6 + q;
                    float top = r0[idx].x + wxa[idx] * (r0[idx].y - r0[idx].x);
                    float bot = r1[idx].x + wxa[idx] * (r1[idx].y - r1[idx].x);
                    acc *= top + wya[idx] * (bot - top);
                }
                // NT store: 384MB write-once stream must not evict L2-resident planes
                __builtin_nontemporal_store(acc, out + (size_t)i * 96 + (size_t)(r * 32) + (size_t)lane);
            }
        }
        __syncthreads();   // protect LDS staging buffer before next chunk
    }
}

extern "C" void kernel_launch(void* const* d_in, const int* in_sizes, int n_in,
                              void* d_out, int out_size, void* d_ws, size_t ws_size,
                              hipStream_t stream) {
    (void)n_in; (void)out_size; (void)d_ws; (void)ws_size;
    const float* pts = (const float*)d_in[0];
    const float* ts  = (const float*)d_in[1];
    PlanePtrs g;
    for (int k = 0; k < 18; ++k) g.p[k] = (const float*)d_in[2 + k];
    const int N = in_sizes[0] / 3;

    const int blocks = 4096;   // persistent-style grid-stride, 8 pts per block-iter
    hexplane_kernel<<<blocks, BLOCK_THREADS, 0, stream>>>(pts, ts, g, (float*)d_out, N);
}